// UniformShardedSNN_89704686944332
// MI455X (gfx1250) — compile-verified
//
#include <hip/hip_runtime.h>
#include <hip/hip_bf16.h>

typedef __attribute__((ext_vector_type(2))) float v2f;
typedef __attribute__((ext_vector_type(8))) float v8f;
typedef __attribute__((address_space(3))) float lds_f;

#define B_SZ    4096
#define T_SZ    32
#define L_SZ    20
#define E_SZ    100000
#define D_SZ    64
#define DENSE_SZ 256
#define HID_OUT 512
#define FEAT_SZ (D_SZ * T_SZ + D_SZ) /* 2112 */

/* ---- WMMA fp32 GEMM tiling ---- */
#define BM 64
#define BN 64
#define BK 32
#define LDA_S 36   /* padded LDS row stride (floats) for A tile */
#define WF_S  160  /* floats per k-pair row of interleaved W tile:
                      160 % 64 == 32 -> lane halves use disjoint LDS bank halves */

// C[M,N] = relu(A[M,K] @ W[K,N] + bias[N]); all row-major.
// Requires M%64==0, N%64==0, K%32==0 (true for all three GEMMs here).
// 256 threads = 8 wave32; waves tiled 4(M) x 2(N); each wave -> 16x32 output.
__global__ __launch_bounds__(256) void gemm_bias_relu(
    const float* __restrict__ A, const float* __restrict__ W,
    const float* __restrict__ bias, float* __restrict__ C,
    int K, int lda, int ldn, int ldc)
{
    __shared__ __attribute__((aligned(16))) float As[BM * LDA_S];
    __shared__ __attribute__((aligned(16))) float Wf[(BK / 2) * WF_S];

    const int tid   = threadIdx.x;
    const int lane  = tid & 31;
    const int wid   = tid >> 5;
    const int waveM = wid >> 1;     /* 0..3 */
    const int waveN = wid & 1;      /* 0..1 */
    const int half  = lane >> 4;    /* 0 = lanes 0-15, 1 = lanes 16-31 */
    const int l15   = lane & 15;

    const int bm = blockIdx.x * BM;
    const int bn = blockIdx.y * BN;

    v8f acc0 = {};
    v8f acc1 = {};

    for (int k0 = 0; k0 < K; k0 += BK) {
        /* ---- Stage A tile (64x32) via async global->LDS DMA: 512 x 16B,
           2 transfers/thread. Per-lane LDS offsets keep the 36-float
           anti-bank-conflict padding. Tracked by ASYNCcnt. ---- */
        #pragma unroll
        for (int p = 0; p < 2; ++p) {
            const int g   = p * 256 + tid;
            const int row = g >> 3;           /* 8 x float4 per row */
            const int c4  = (g & 7) * 4;
            const float* gp = A + (size_t)(bm + row) * lda + k0 + c4;
            const unsigned loff =
                (unsigned)(unsigned long long)(lds_f*)(As + row * LDA_S + c4);
            asm volatile("global_load_async_to_lds_b128 %0, %1, off"
                         :: "v"(loff), "v"(gp) : "memory");
        }

        /* ---- Stage W tile (32x64) into k-pair interleaved, fragment-ready
           layout: Wf[kp*WF_S + n*2 + j] = W[k0+2*kp+j][bn+n].
           256 threads cover 16 kpairs x 16 col-quads exactly. ---- */
        {
            const int kp = tid >> 4;          /* 0..15 */
            const int nq = (tid & 15) * 4;    /* 0..60 */
            const float* w0 = W + (size_t)(k0 + 2 * kp) * ldn + bn + nq;
            const float4 r0 = *(const float4*)w0;
            const float4 r1 = *(const float4*)(w0 + ldn);
            float* d = Wf + kp * WF_S + nq * 2;
            *(float4*)(d)     = make_float4(r0.x, r1.x, r0.y, r1.y);
            *(float4*)(d + 4) = make_float4(r0.z, r1.z, r0.w, r1.w);
        }

        asm volatile("s_wait_asynccnt 0x0" ::: "memory");
        __syncthreads();

        const int arow = waveM * 16 + l15;
        #pragma unroll
        for (int kk = 0; kk < BK / 4; ++kk) {
            /* A fragment 16x4: lanes 0-15 hold K=kb..kb+1, lanes 16-31 kb+2..kb+3 */
            const int kb = kk * 4 + half * 2;
            const v2f a = *(const v2f*)(As + arow * LDA_S + kb);
            /* B fragments: one aligned ds_load_b64 each from interleaved layout */
            const int kprow = (kk * 2 + half) * WF_S;
            {
                const v2f b = *(const v2f*)(Wf + kprow + (waveN * 32 + l15) * 2);
                acc0 = __builtin_amdgcn_wmma_f32_16x16x4_f32(
                    false, a, false, b, (short)0, acc0, false, false);
            }
            {
                const v2f b = *(const v2f*)(Wf + kprow + (waveN * 32 + 16 + l15) * 2);
                acc1 = __builtin_amdgcn_wmma_f32_16x16x4_f32(
                    false, a, false, b, (short)0, acc1, false, false);
            }
        }
        __syncthreads();
    }

    /* Epilogue: bias + relu + store.
       D layout: VGPR i -> row = i + 8*half, col = l15 (per 16x16 subtile). */
    const int col0 = bn + waveN * 32 + l15;
    const float b0 = bias[col0];
    const float b1 = bias[col0 + 16];
    #pragma unroll
    for (int i = 0; i < 8; ++i) {
        const int row = bm + waveM * 16 + half * 8 + i;
        C[(size_t)row * ldc + col0]      = fmaxf(acc0[i] + b0, 0.0f);
        C[(size_t)row * ldc + col0 + 16] = fmaxf(acc1[i] + b1, 0.0f);
    }
}

/* ---- Embedding-bag sum pooling: one wave per (b, t) pair ----
   Lane holds 2 of the 64 row floats -> each row read = 256B fully coalesced.
   Prefetch next row while accumulating current one (global_prefetch_b8). */
__global__ __launch_bounds__(256) void emb_pool(
    const int* __restrict__ idx, const float* __restrict__ tables,
    float* __restrict__ feat)
{
    const int gw   = (blockIdx.x * 256 + threadIdx.x) >> 5; /* global wave id */
    const int lane = threadIdx.x & 31;
    const int b = gw / T_SZ;
    const int t = gw % T_SZ;

    const int* ip = idx + ((size_t)b * T_SZ + t) * L_SZ;
    const float* tab = tables + (size_t)t * E_SZ * D_SZ;

    float2 acc = make_float2(0.0f, 0.0f);
    int cur = ip[0];
    for (int l = 0; l < L_SZ; ++l) {
        const float* row = tab + (size_t)cur * D_SZ + 2 * lane;
        if (l + 1 < L_SZ) {
            const int nxt = ip[l + 1];
            __builtin_prefetch(tab + (size_t)nxt * D_SZ + 2 * lane, 0, 0);
            cur = nxt;
        }
        const float2 v = *(const float2*)row;
        acc.x += v.x;
        acc.y += v.y;
    }
    float2* dst = (float2*)(feat + (size_t)b * FEAT_SZ + D_SZ + t * D_SZ + 2 * lane);
    *dst = acc;
}

/* ---- Output fc2: [4096,512] x [512,1] matvec + bias + relu; one wave/row */
__global__ __launch_bounds__(256) void out_fc2(
    const float* __restrict__ H, const float* __restrict__ w,
    const float* __restrict__ bptr, float* __restrict__ out)
{
    const int gw   = (blockIdx.x * 256 + threadIdx.x) >> 5;
    const int lane = threadIdx.x & 31;
    const float* h = H + (size_t)gw * HID_OUT;
    float s = 0.0f;
    #pragma unroll
    for (int j = 0; j < HID_OUT / 32; ++j)
        s += h[lane + 32 * j] * w[lane + 32 * j];
    #pragma unroll
    for (int o = 16; o > 0; o >>= 1)
        s += __shfl_xor(s, o, 32);
    if (lane == 0)
        out[gw] = fmaxf(s + bptr[0], 0.0f);
}

extern "C" void kernel_launch(void* const* d_in, const int* in_sizes, int n_in,
                              void* d_out, int out_size, void* d_ws, size_t ws_size,
                              hipStream_t stream) {
    const float* dense  = (const float*)d_in[0];
    const int*   sparse = (const int*)  d_in[1];
    const float* tables = (const float*)d_in[2];
    const float* dw1 = (const float*)d_in[3];
    const float* db1 = (const float*)d_in[4];
    const float* dw2 = (const float*)d_in[5];
    const float* db2 = (const float*)d_in[6];
    const float* ow1 = (const float*)d_in[7];
    const float* ob1 = (const float*)d_in[8];
    const float* ow2 = (const float*)d_in[9];
    const float* ob2 = (const float*)d_in[10];
    float* out = (float*)d_out;

    float* feat = (float*)d_ws;                         /* [B, 2112] */
    float* h1   = feat + (size_t)B_SZ * FEAT_SZ;        /* [B, 256]  */
    float* hout = h1   + (size_t)B_SZ * DENSE_SZ;       /* [B, 512]  */

    /* 1) embedding-bag pooling -> feat[:, 64:2112] */
    emb_pool<<<dim3((B_SZ * T_SZ) / 8), 256, 0, stream>>>(sparse, tables, feat);

    /* 2) dense fc1: [4096,256]x[256,256] -> h1 */
    gemm_bias_relu<<<dim3(B_SZ / BM, DENSE_SZ / BN), 256, 0, stream>>>(
        dense, dw1, db1, h1, DENSE_SZ, DENSE_SZ, DENSE_SZ, DENSE_SZ);

    /* 3) dense fc2: [4096,256]x[256,64] -> feat[:, 0:64] (ldc = 2112) */
    gemm_bias_relu<<<dim3(B_SZ / BM, D_SZ / BN), 256, 0, stream>>>(
        h1, dw2, db2, feat, DENSE_SZ, DENSE_SZ, D_SZ, FEAT_SZ);

    /* 4) output fc1: [4096,2112]x[2112,512] -> hout */
    gemm_bias_relu<<<dim3(B_SZ / BM, HID_OUT / BN), 256, 0, stream>>>(
        feat, ow1, ob1, hout, FEAT_SZ, FEAT_SZ, HID_OUT, HID_OUT);

    /* 5) output fc2: [4096,512]x[512,1] -> out */
    out_fc2<<<dim3(B_SZ / 8), 256, 0, stream>>>(hout, ow2, ob2, out);
}